// Encoder_24919400252010
// MI455X (gfx1250) — compile-verified
//
#include <hip/hip_runtime.h>
#include <hip/hip_bf16.h>

typedef float v2f __attribute__((ext_vector_type(2)));
typedef float v8f __attribute__((ext_vector_type(8)));

#define N_NODES 100000
#define N_EDGES 3200000
#define XF      7      // IN_CH - 1
#define EMBD    192
#define K0      200    // 199 padded to 200 for K%4==0
#define AP      202    // LDS row pitch: even (8B-aligned pairs) + conflict-free (202%64=10)
#define HID     32
#define OUTF    16
#define ROWS_PB 128    // rows per block (8 waves x 16)

// ---------------- degree / norm ----------------
__global__ void k_deg_init(float* __restrict__ deg) {
    int i = blockIdx.x * 256 + threadIdx.x;
    if (i < N_NODES) deg[i] = 1.0f;               // self loop
}

__global__ void k_deg_count(const int* __restrict__ dst, float* __restrict__ deg) {
    int e = blockIdx.x * 256 + threadIdx.x;
    if (e < N_EDGES) atomicAdd(&deg[dst[e]], 1.0f);
}

__global__ void k_dinv(float* __restrict__ dinv) {
    int i = blockIdx.x * 256 + threadIdx.x;
    if (i < N_NODES) dinv[i] = 1.0f / sqrtf(dinv[i]);
}

__global__ void k_norm(const int* __restrict__ src, const int* __restrict__ dst,
                       const float* __restrict__ dinv, float* __restrict__ norm) {
    int e = blockIdx.x * 256 + threadIdx.x;
    if (e < N_EDGES) norm[e] = dinv[src[e]] * dinv[dst[e]];
}

// ---------------- layer-0 GEMM: concat(x_feat, emb[ids])[N,199] @ W0[199,32] -> h, agg ----------------
// A tile staged to LDS once (concat logic outside the K loop); K loop is pure LDS + WMMA.
// Self-loop term fused into epilogue: agg = acc * dinv^2.
__global__ __launch_bounds__(256)
void k_gemm0(const float* __restrict__ xfeat, const int* __restrict__ nodeids,
             const float* __restrict__ emb, const float* __restrict__ W0,
             const float* __restrict__ dinv,
             float* __restrict__ h, float* __restrict__ agg) {
    __shared__ int   nlds[ROWS_PB];
    __shared__ float alds[ROWS_PB * AP];          // ~101 KB
    __shared__ float wlds[K0 * HID];              // 25.6 KB, row 199 zero-padded

    int tid = threadIdx.x;
    int blk_row0 = blockIdx.x * ROWS_PB;

    if (tid < ROWS_PB)
        nlds[tid] = nodeids[min(blk_row0 + tid, N_NODES - 1)];
    __syncthreads();

    // stage weights (zero-pad K row 199)
    for (int idx = tid; idx < K0 * HID; idx += 256) {
        int k = idx >> 5;
        wlds[idx] = (k < K0 - 1) ? W0[idx] : 0.0f;
    }
    // stage A tile: cols 0..6 from x_feat, 7..198 from emb (emb[0] is zeros = padding_idx)
    for (int idx = tid; idx < ROWS_PB * (K0 - 1); idx += 256) {
        int r = idx / (K0 - 1), c = idx - r * (K0 - 1);   // c in [0,198]
        int grow = min(blk_row0 + r, N_NODES - 1);
        float v = (c < XF) ? xfeat[(size_t)grow * XF + c]
                           : emb[(size_t)nlds[r] * EMBD + (c - XF)];
        alds[r * AP + c] = v;
    }
    if (tid < ROWS_PB) alds[tid * AP + (K0 - 1)] = 0.0f; // pad col 199
    __syncthreads();

    int wave = tid >> 5, lane = tid & 31;
    int m = lane & 15, kh = lane >> 4;            // A row-in-tile, K-half select
    const float* aw = alds + (wave * 16) * AP;

    v8f acc0 = {}, acc1 = {};
    for (int kb = 0; kb < K0; kb += 4) {
        int k = kb + 2 * kh;                      // this lane's K pair (even)
        v2f a;  a.x  = aw[m * AP + k];            a.y  = aw[m * AP + k + 1];
        v2f b0; b0.x = wlds[k * HID + m];         b0.y = wlds[(k + 1) * HID + m];
        v2f b1; b1.x = wlds[k * HID + 16 + m];    b1.y = wlds[(k + 1) * HID + 16 + m];
        acc0 = __builtin_amdgcn_wmma_f32_16x16x4_f32(false, a, false, b0, (short)0, acc0, false, false);
        acc1 = __builtin_amdgcn_wmma_f32_16x16x4_f32(false, a, false, b1, (short)0, acc1, false, false);
    }

    int row0 = blk_row0 + wave * 16;
    #pragma unroll
    for (int v = 0; v < 8; ++v) {
        int r = row0 + v + 8 * kh;                // D layout: vgpr v -> M = v + 8*(lane/16)
        if (r < N_NODES) {
            float di = dinv[r]; float d2 = di * di;
            size_t o = (size_t)r * HID + m;
            h[o]        = acc0[v];  agg[o]        = acc0[v] * d2;
            h[o + 16]   = acc1[v];  agg[o + 16]   = acc1[v] * d2;
        }
    }
}

// ---------------- hidden GEMM: relu(Ain+bias)[N,32] @ W[32,16*NT] -> h, agg ----------------
template <int NT>
__global__ __launch_bounds__(256)
void k_gemm_hidden(const float* __restrict__ Ain, const float* __restrict__ bin,
                   const float* __restrict__ W, const float* __restrict__ dinv,
                   float* __restrict__ h, float* __restrict__ agg) {
    constexpr int F = 16 * NT;
    __shared__ float wlds[HID * F];
    __shared__ float blds[HID];
    int tid = threadIdx.x;
    for (int idx = tid; idx < HID * F; idx += 256) wlds[idx] = W[idx];
    if (tid < HID) blds[tid] = bin[tid];
    __syncthreads();

    int wave = tid >> 5, lane = tid & 31;
    int m = lane & 15, kh = lane >> 4;
    int row0 = (blockIdx.x * 8 + wave) * 16;
    int rowc = min(row0 + m, N_NODES - 1);        // clamp: garbage only hits unstored rows
    const v2f* arv = (const v2f*)(Ain + (size_t)rowc * HID);

    v8f acc[NT] = {};
    for (int kb = 0; kb < HID; kb += 4) {
        int k = kb + 2 * kh;
        v2f av = arv[k >> 1];                     // fused bias + ReLU of previous layer
        v2f a;  a.x = fmaxf(av.x + blds[k], 0.0f);
                a.y = fmaxf(av.y + blds[k + 1], 0.0f);
        #pragma unroll
        for (int t = 0; t < NT; ++t) {
            v2f b; b.x = wlds[k * F + 16 * t + m]; b.y = wlds[(k + 1) * F + 16 * t + m];
            acc[t] = __builtin_amdgcn_wmma_f32_16x16x4_f32(false, a, false, b, (short)0, acc[t], false, false);
        }
    }
    #pragma unroll
    for (int v = 0; v < 8; ++v) {
        int r = row0 + v + 8 * kh;
        if (r < N_NODES) {
            float di = dinv[r]; float d2 = di * di;
            #pragma unroll
            for (int t = 0; t < NT; ++t) {
                size_t o = (size_t)r * F + 16 * t + m;
                h[o] = acc[t][v];  agg[o] = acc[t][v] * d2;
            }
        }
    }
}

// ---------------- edge scatter: agg[dst] += h[src] * norm ----------------
// lane = feature -> coalesced gather + contiguous no-return f32 atomics (L2-resident)
template <int F>
__global__ void k_scatter(const int* __restrict__ src, const int* __restrict__ dst,
                          const float* __restrict__ norm, const float* __restrict__ h,
                          float* __restrict__ agg) {
    long long t = (long long)blockIdx.x * 256 + threadIdx.x;
    int e = (int)(t / F);
    int j = (int)(t & (F - 1));
    if (e < N_EDGES) {
        int s = src[e], d = dst[e];
        float v = h[(size_t)s * F + j] * norm[e];
        atomicAdd(&agg[(size_t)d * F + j], v);
    }
}

// ---------------- final bias (no relu) ----------------
__global__ void k_out(const float* __restrict__ agg, const float* __restrict__ bmu,
                      float* __restrict__ out) {
    int i = blockIdx.x * 256 + threadIdx.x;
    if (i < N_NODES * OUTF) out[i] = agg[i] + bmu[i & (OUTF - 1)];
}

static inline int cdiv(long long a, long long b) { return (int)((a + b - 1) / b); }

extern "C" void kernel_launch(void* const* d_in, const int* in_sizes, int n_in,
                              void* d_out, int out_size, void* d_ws, size_t ws_size,
                              hipStream_t stream) {
    const float* xfeat   = (const float*)d_in[0];
    const int*   nodeids = (const int*)  d_in[1];
    const int*   ei      = (const int*)  d_in[2];
    const float* emb     = (const float*)d_in[3];
    const float* W0  = (const float*)d_in[4];
    const float* b0  = (const float*)d_in[5];
    const float* W1  = (const float*)d_in[6];
    const float* b1  = (const float*)d_in[7];
    const float* W2  = (const float*)d_in[8];
    const float* b2  = (const float*)d_in[9];
    const float* Wmu = (const float*)d_in[10];
    const float* bmu = (const float*)d_in[11];
    float* out = (float*)d_out;

    const int* src = ei;                // edge_index[0]
    const int* dst = ei + N_EDGES;      // edge_index[1]

    // workspace: dinv[N] | norm[E] | h[N*32] | agg[N*32]  (~39 MB)
    float* dinv = (float*)d_ws;
    float* norm = dinv + N_NODES;
    float* h    = norm + N_EDGES;
    float* agg  = h + (size_t)N_NODES * HID;

    // degree + symmetric normalization
    k_deg_init <<<cdiv(N_NODES, 256), 256, 0, stream>>>(dinv);
    k_deg_count<<<cdiv(N_EDGES, 256), 256, 0, stream>>>(dst, dinv);
    k_dinv     <<<cdiv(N_NODES, 256), 256, 0, stream>>>(dinv);
    k_norm     <<<cdiv(N_EDGES, 256), 256, 0, stream>>>(src, dst, dinv, norm);

    const int gemm_blocks = cdiv(N_NODES, ROWS_PB);

    // layer 0
    k_gemm0<<<gemm_blocks, 256, 0, stream>>>(xfeat, nodeids, emb, W0, dinv, h, agg);
    k_scatter<HID><<<cdiv((long long)N_EDGES * HID, 256), 256, 0, stream>>>(src, dst, norm, h, agg);
    // layer 1
    k_gemm_hidden<2><<<gemm_blocks, 256, 0, stream>>>(agg, b0, W1, dinv, h, agg);
    k_scatter<HID><<<cdiv((long long)N_EDGES * HID, 256), 256, 0, stream>>>(src, dst, norm, h, agg);
    // layer 2
    k_gemm_hidden<2><<<gemm_blocks, 256, 0, stream>>>(agg, b1, W2, dinv, h, agg);
    k_scatter<HID><<<cdiv((long long)N_EDGES * HID, 256), 256, 0, stream>>>(src, dst, norm, h, agg);
    // mu layer (Fout = 16, final bias added after aggregation, no relu)
    k_gemm_hidden<1><<<gemm_blocks, 256, 0, stream>>>(agg, b2, Wmu, dinv, h, agg);
    k_scatter<OUTF><<<cdiv((long long)N_EDGES * OUTF, 256), 256, 0, stream>>>(src, dst, norm, h, agg);
    k_out<<<cdiv((long long)N_NODES * OUTF, 256), 256, 0, stream>>>(agg, bmu, out);
}